// LMMD_loss_46497315947051
// MI455X (gfx1250) — compile-verified
//
#include <hip/hip_runtime.h>
#include <math.h>
#include <stdint.h>

typedef __attribute__((ext_vector_type(2))) float v2f;
typedef __attribute__((ext_vector_type(8))) float v8f;
typedef unsigned int u32x4 __attribute__((ext_vector_type(4)));
typedef int i32x4 __attribute__((ext_vector_type(4)));
typedef int i32x8 __attribute__((ext_vector_type(8)));

#define NROWS 4096
#define DDIM  256
#define CCLS  31
#define NTOT  8192
#define NTIL  256            // 4096/16 tiles per dimension
#define KCH   128            // K-chunk staged in LDS
#define RS    132            // padded LDS row stride (floats): conflict-free fragment reads

// workspace layout (float indices)
#define OFF_BW    0
#define OFF_SCALE 1
#define OFF_SINV  8                   // 32 floats (c=31 padded 0)
#define OFF_TINV  40                  // 32 floats
#define OFF_SQ    128                 // 8192 floats: sq_s[4096] then sq_t[4096]
#define OFF_SV    (OFF_SQ + NTOT)     // 4096*32
#define OFF_TV    (OFF_SV + NROWS*32) // 4096*32
#define OFF_PART  (OFF_TV + NROWS*32) // 8192 block partials

// LDS: two buffers (double-buffered K-chunks); each: sI tI (16 rows) sJ tJ (128 rows)
#define L_SI 0
#define L_TI (16*RS)
#define L_SJ (32*RS)
#define L_TJ (160*RS)
#define BUF  (288*RS)                 // floats per buffer
#define LDS_FLOATS (2*BUF)            // 76032 floats = 304128 bytes (<320KB WGP LDS)

// ---------------- pass 1a: per-row squared norms ----------------
__global__ void k_row_sq(const float* __restrict__ src, const float* __restrict__ tgt,
                         float* __restrict__ ws) {
  int i = blockIdx.x * blockDim.x + threadIdx.x;
  if (i >= NTOT) return;
  const float* row = (i < NROWS) ? (src + (size_t)i * DDIM)
                                 : (tgt + (size_t)(i - NROWS) * DDIM);
  float s = 0.f;
  #pragma unroll 8
  for (int d = 0; d < DDIM; d += 4) {
    float4 v = *(const float4*)(row + d);
    s += v.x * v.x + v.y * v.y + v.z * v.z + v.w * v.w;
  }
  ws[OFF_SQ + i] = s;
}

// ---------------- pass 1b: bandwidth via  sum(d2) = 2n*sumSq - 2*||colsum||^2 ----
__global__ void k_bandwidth(const float* __restrict__ src, const float* __restrict__ tgt,
                            float* __restrict__ ws) {
  __shared__ float red[256];
  int t = threadIdx.x;                      // 256 threads, one per dim
  float cs = 0.f;
  for (int i = 0; i < NROWS; ++i) cs += src[(size_t)i * DDIM + t];
  for (int i = 0; i < NROWS; ++i) cs += tgt[(size_t)i * DDIM + t];
  float csq = cs * cs;

  float ss = 0.f;
  for (int k = 0; k < NTOT / 256; ++k) ss += ws[OFF_SQ + t + k * 256];

  red[t] = csq; __syncthreads();
  for (int o = 128; o > 0; o >>= 1) { if (t < o) red[t] += red[t + o]; __syncthreads(); }
  float sumG = red[0]; __syncthreads();
  red[t] = ss; __syncthreads();
  for (int o = 128; o > 0; o >>= 1) { if (t < o) red[t] += red[t + o]; __syncthreads(); }
  if (t == 0) {
    double sumSq = (double)red[0];
    double n = (double)NTOT;
    double sumd2 = 2.0 * n * sumSq - 2.0 * (double)sumG;
    double bw = sumd2 / (n * n - n);
    bw = bw / 4.0;                    // KERNEL_MUL ** (KERNEL_NUM // 2)
    ws[OFF_BW] = (float)bw;
  }
}

// ---------------- pass 2a: class stats -> inverse denominators, mask, scale ----
__global__ void k_weight_stats(const int* __restrict__ s_label, const float* __restrict__ t_label,
                               float* __restrict__ ws) {
  __shared__ int cnt[32];
  __shared__ int pres[32];
  __shared__ int msk[32];
  int t = threadIdx.x;                      // 256 threads
  if (t < 32) { cnt[t] = 0; pres[t] = 0; msk[t] = 0; }
  __syncthreads();
  for (int i = t; i < NROWS; i += 256) {
    int lab = s_label[i];
    if (lab >= 0 && lab < CCLS) atomicAdd(&cnt[lab], 1);
    const float* r = t_label + (size_t)i * CCLS;
    float best = r[0]; int bi = 0;
    for (int c = 1; c < CCLS; ++c) { float v = r[c]; if (v > best) { best = v; bi = c; } }
    atomicOr(&pres[bi], 1);
  }
  __syncthreads();
  if (t < 32) {
    if (t < CCLS) {
      float tsum = 0.f;
      for (int i = 0; i < NROWS; ++i) tsum += t_label[(size_t)i * CCLS + t];
      int m = (cnt[t] > 0) && pres[t];
      msk[t] = m;
      float sden = (cnt[t] == 0) ? 100.0f : (float)cnt[t];
      float tden = (tsum == 0.0f) ? 100.0f : tsum;
      ws[OFF_SINV + t] = m ? (1.0f / sden) : 0.0f;
      ws[OFF_TINV + t] = m ? (1.0f / tden) : 0.0f;
    } else {
      ws[OFF_SINV + t] = 0.0f;
      ws[OFF_TINV + t] = 0.0f;
    }
  }
  __syncthreads();
  if (t == 0) {
    int len = 0;
    for (int c = 0; c < CCLS; ++c) len += msk[c];
    ws[OFF_SCALE] = (len > 0) ? (1.0f / (float)len) : 0.0f;
  }
}

// ---------------- pass 2b: fill s_vec / t_vec (4096 x 32, col 31 zero-padded) ---
__global__ void k_fill_vecs(const int* __restrict__ s_label, const float* __restrict__ t_label,
                            float* __restrict__ ws) {
  int i = blockIdx.x * blockDim.x + threadIdx.x;
  if (i >= NROWS) return;
  int lab = s_label[i];
  float* sv = ws + OFF_SV + (size_t)i * 32;
  float* tv = ws + OFF_TV + (size_t)i * 32;
  const float* tr = t_label + (size_t)i * CCLS;
  #pragma unroll
  for (int c = 0; c < 32; ++c) {
    float sval = (c < CCLS && lab == c) ? ws[OFF_SINV + c] : 0.0f;
    float tval = (c < CCLS) ? tr[c] * ws[OFF_TINV + c] : 0.0f;
    sv[c] = sval;
    tv[c] = tval;
  }
}

// ---- TDM: one 2D tile (nrows x 128 floats, row stride 256) -> padded LDS rows
// pad_interval=6 (after 128 DWORDs) + pad_amount=3 (4 DWORDs) reproduces RS=132.
__device__ __forceinline__ void tdm_load_tile(const float* gptr, int nrows, uint32_t lds_byte) {
  uint64_t ga = (uint64_t)(uintptr_t)gptr;
  u32x4 g0;
  g0.x = 1u;                                              // count=1 (valid descriptor)
  g0.y = lds_byte;                                        // lds_addr
  g0.z = (uint32_t)ga;                                    // global_addr[31:0] (bits 95:64)
  g0.w = (uint32_t)((ga >> 32) & 0x1FFFFFFu) | (2u << 30); // global_addr[56:32] | type=2
  i32x8 g1;
  g1[0] = (2 << 16) | (1 << 20) | (6 << 22) | (3 << 25);  // data_size=4B, pad_en, int=128dw, amt=4dw
  g1[1] = (int)(256u << 16);                              // tensor_dim0=256 (lo16 @ [31:16])
  g1[2] = (int)(4096u << 16);                             // dim0 hi | tensor_dim1=4096 lo16
  g1[3] = (int)(128u << 16);                              // dim1 hi | tile_dim0=128
  g1[4] = nrows;                                          // tile_dim1 (tile_dim2=0)
  g1[5] = 256;                                            // tensor_dim0_stride lo32
  g1[6] = 0;                                              // stride hi | dim1_stride lo
  g1[7] = 0;
  i32x4 z4 = {0, 0, 0, 0};
#if defined(__clang_major__) && __clang_major__ >= 23
  i32x8 z8 = {0, 0, 0, 0, 0, 0, 0, 0};
  __builtin_amdgcn_tensor_load_to_lds(g0, g1, z4, z4, z8, 0);
#else
  __builtin_amdgcn_tensor_load_to_lds(g0, g1, z4, z4, 0);
#endif
}

// ---------------- pass 3: fused Gram + weights + gaussian + weighted reduce -----
// block = 8 waves; owns row-tile ti and 8 consecutive col-tiles tj0..tj0+7.
// Double-buffered K-chunks staged by the Tensor Data Mover; WMMA fragments read
// from LDS bank-conflict-free (RS=132: 32 lanes cover all 64 banks once).
#define GRAM_CHUNK(BASE)                                                          \
  {                                                                               \
    const float* aSp = lds + (BASE) + L_SI + m * RS + koff;                       \
    const float* aTp = lds + (BASE) + L_TI + m * RS + koff;                       \
    const float* bSp = lds + (BASE) + L_SJ + (wave * 16 + m) * RS + koff;         \
    const float* bTp = lds + (BASE) + L_TJ + (wave * 16 + m) * RS + koff;         \
    _Pragma("unroll 8")                                                           \
    for (int k = 0; k < KCH; k += 4) {                                            \
      v2f aS = *(const v2f*)(aSp + k);                                            \
      v2f bS = *(const v2f*)(bSp + k);                                            \
      v2f aT = *(const v2f*)(aTp + k);                                            \
      v2f bT = *(const v2f*)(bTp + k);                                            \
      gSS = __builtin_amdgcn_wmma_f32_16x16x4_f32(false, aS, false, bS, (short)0, gSS, false, false); \
      gTT = __builtin_amdgcn_wmma_f32_16x16x4_f32(false, aT, false, bT, (short)0, gTT, false, false); \
      gST = __builtin_amdgcn_wmma_f32_16x16x4_f32(false, aS, false, bT, (short)0, gST, false, false); \
    }                                                                             \
  }

__global__ void __launch_bounds__(256) k_lmmd_main(const float* __restrict__ src,
                                                   const float* __restrict__ tgt,
                                                   float* __restrict__ ws) {
  extern __shared__ float lds[];
  const int tid  = threadIdx.x;
  const int wave = tid >> 5;
  const int lane = tid & 31;
  const int ti  = blockIdx.x >> 5;        // row tile 0..255
  const int tj0 = (blockIdx.x & 31) * 8;  // first col tile of this block's strip
  const int tj  = tj0 + wave;             // this wave's col tile
  const int m    = lane & 15;             // fragment row within tile
  const int half = lane >> 4;
  const int koff = half * 2;              // f32 WMMA A/B K-offset for upper half-wave

  // ---- wave 0 issues all 8 TDM descriptors (both chunks, both buffers) up front
  if (wave == 0) {
    const float* sI = src + (size_t)(ti * 16) * DDIM;
    const float* tI = tgt + (size_t)(ti * 16) * DDIM;
    const float* sJ = src + (size_t)(tj0 * 16) * DDIM;
    const float* tJ = tgt + (size_t)(tj0 * 16) * DDIM;
    // chunk 0 -> buffer 0
    tdm_load_tile(sI,        16,  (uint32_t)(uintptr_t)(lds + 0 * BUF + L_SI));
    tdm_load_tile(tI,        16,  (uint32_t)(uintptr_t)(lds + 0 * BUF + L_TI));
    tdm_load_tile(sJ,        128, (uint32_t)(uintptr_t)(lds + 0 * BUF + L_SJ));
    tdm_load_tile(tJ,        128, (uint32_t)(uintptr_t)(lds + 0 * BUF + L_TJ));
    // chunk 1 -> buffer 1 (overlaps with chunk-0 compute)
    tdm_load_tile(sI + KCH,  16,  (uint32_t)(uintptr_t)(lds + 1 * BUF + L_SI));
    tdm_load_tile(tI + KCH,  16,  (uint32_t)(uintptr_t)(lds + 1 * BUF + L_TI));
    tdm_load_tile(sJ + KCH,  128, (uint32_t)(uintptr_t)(lds + 1 * BUF + L_SJ));
    tdm_load_tile(tJ + KCH,  128, (uint32_t)(uintptr_t)(lds + 1 * BUF + L_TJ));
    __builtin_amdgcn_s_wait_tensorcnt(4);   // in-order: first 4 (chunk 0) complete
  }
  __syncthreads();

  v8f gSS = {}; v8f gTT = {}; v8f gST = {};
  GRAM_CHUNK(0 * BUF)                       // compute chunk 0 while TDM streams chunk 1

  if (wave == 0) __builtin_amdgcn_s_wait_tensorcnt(0);
  __syncthreads();
  GRAM_CHUNK(1 * BUF)

  // ---- rank-31 weight Grams (K=32 padded), small: straight from global/L2
  const float* svA = ws + OFF_SV + (size_t)(ti * 16 + m) * 32 + koff;
  const float* svB = ws + OFF_SV + (size_t)(tj * 16 + m) * 32 + koff;
  const float* tvA = ws + OFF_TV + (size_t)(ti * 16 + m) * 32 + koff;
  const float* tvB = ws + OFF_TV + (size_t)(tj * 16 + m) * 32 + koff;
  v8f wSS = {}; v8f wTT = {}; v8f wST = {};
  #pragma unroll
  for (int k = 0; k < 32; k += 4) {
    v2f aS = *(const v2f*)(svA + k);
    v2f bS = *(const v2f*)(svB + k);
    v2f aT = *(const v2f*)(tvA + k);
    v2f bT = *(const v2f*)(tvB + k);
    wSS = __builtin_amdgcn_wmma_f32_16x16x4_f32(false, aS, false, bS, (short)0, wSS, false, false);
    wTT = __builtin_amdgcn_wmma_f32_16x16x4_f32(false, aT, false, bT, (short)0, wTT, false, false);
    wST = __builtin_amdgcn_wmma_f32_16x16x4_f32(false, aS, false, bT, (short)0, wST, false, false);
  }

  // ---- epilogue: d2 -> 5-exponential gaussian sum, weighted accumulate
  const float g0 = 1.0f / ws[OFF_BW];
  const int   ncol = tj * 16 + (lane & 15);        // D layout: N = lane&15
  const float sq_s_j = ws[OFF_SQ + ncol];
  const float sq_t_j = ws[OFF_SQ + NROWS + ncol];
  float part = 0.f;
  #pragma unroll
  for (int r = 0; r < 8; ++r) {
    int grow = ti * 16 + half * 8 + r;             // D layout: M = r + 8*half
    float sq_s_i = ws[OFF_SQ + grow];
    float sq_t_i = ws[OFF_SQ + NROWS + grow];
    float dss = sq_s_i + sq_s_j - 2.0f * gSS[r];
    float dtt = sq_t_i + sq_t_j - 2.0f * gTT[r];
    float dst = sq_s_i + sq_t_j - 2.0f * gST[r];
    float kss = 0.f, ktt = 0.f, kst = 0.f;
    float g = g0;
    #pragma unroll
    for (int q = 0; q < 5; ++q) {
      kss += __expf(-dss * g);
      ktt += __expf(-dtt * g);
      kst += __expf(-dst * g);
      g *= 0.5f;
    }
    part += wSS[r] * kss + wTT[r] * ktt - 2.0f * wST[r] * kst;
  }

  // fixed-order wave reduction (wave32), then fixed-order block reduction
  for (int off = 16; off > 0; off >>= 1) part += __shfl_xor(part, off, 32);
  __shared__ float sp[8];
  if (lane == 0) sp[wave] = part;
  __syncthreads();
  if (tid == 0) {
    float s = 0.f;
    for (int w = 0; w < 8; ++w) s += sp[w];
    ws[OFF_PART + blockIdx.x] = s;
  }
}

// ---------------- pass 4: final deterministic reduction, scale, nan guard ------
__global__ void k_finalize(const float* __restrict__ ws, float* __restrict__ out) {
  __shared__ float red[256];
  int t = threadIdx.x;
  float s = 0.f;
  for (int k = 0; k < 8192 / 256; ++k) s += ws[OFF_PART + t + k * 256];
  red[t] = s; __syncthreads();
  for (int o = 128; o > 0; o >>= 1) { if (t < o) red[t] += red[t + o]; __syncthreads(); }
  if (t == 0) {
    float loss = red[0] * ws[OFF_SCALE];
    if (__builtin_isnan(loss)) loss = 0.f;
    out[0] = loss;
  }
}

extern "C" void kernel_launch(void* const* d_in, const int* in_sizes, int n_in,
                              void* d_out, int out_size, void* d_ws, size_t ws_size,
                              hipStream_t stream) {
  const float* src     = (const float*)d_in[0];
  const float* tgt     = (const float*)d_in[1];
  const int*   s_label = (const int*)d_in[2];
  const float* t_label = (const float*)d_in[3];
  float* ws  = (float*)d_ws;
  float* out = (float*)d_out;

  k_row_sq      <<<NTOT / 256, 256, 0, stream>>>(src, tgt, ws);
  k_bandwidth   <<<1,          256, 0, stream>>>(src, tgt, ws);
  k_weight_stats<<<1,          256, 0, stream>>>(s_label, t_label, ws);
  k_fill_vecs   <<<NROWS / 256,256, 0, stream>>>(s_label, t_label, ws);
  k_lmmd_main   <<<NTIL * NTIL / 8, 256, LDS_FLOATS * sizeof(float), stream>>>(src, tgt, ws);
  k_finalize    <<<1,          256, 0, stream>>>(ws, out);
}